// QLSTMCell_25142738551335
// MI455X (gfx1250) — compile-verified
//
#include <hip/hip_runtime.h>
#include <hip/hip_bf16.h>

#define NQ      14
#define NSTATE  (1 << NQ)       // 16384 amplitudes
#define NLAYERS 2
#define BATCH   256
#define HID     8
#define INSZ    6

typedef float v2f __attribute__((ext_vector_type(2)));
typedef float v8f __attribute__((ext_vector_type(8)));

// ---------- complex helpers (float2 = {re, im}) ----------
__device__ __forceinline__ float2 cmul(float2 a, float2 b) {
    return make_float2(a.x * b.x - a.y * b.y, a.x * b.y + a.y * b.x);
}
__device__ __forceinline__ float2 cadd(float2 a, float2 b) {
    return make_float2(a.x + b.x, a.y + b.y);
}

// =====================================================================
// Kernel 1: full 14-qubit state-vector circuit per batch element.
// One workgroup per batch element; entire 128KB state lives in LDS
// (CDNA5 WGP has 320KB LDS -> 2 workgroups/WGP). Zero HBM traffic for
// the state: initial product state is built analytically, all 56 gates
// applied in LDS, only 14 expectation values leave the WGP.
// =====================================================================
__global__ __launch_bounds__(256) void qcircuit_kernel(
    const float* __restrict__ x,     // [256, 6]
    const float* __restrict__ h,     // [256, 8]
    const float* __restrict__ wts,   // [2, 14, 3]
    float* __restrict__ Qws)         // [256, 16]  (q padded K 14->16)
{
    extern __shared__ float2 st[];                 // NSTATE float2 = 128KB
    float* red = (float*)(st + NSTATE);            // 16 floats for reduction

    const int b   = blockIdx.x;
    const int tid = threadIdx.x;

    // ---- per-qubit half-angle cos/sin (combined = concat(x, h)) ----
    float ch[NQ], sh[NQ];
#pragma unroll
    for (int w = 0; w < NQ; ++w) {
        float ang  = (w < INSZ) ? x[b * INSZ + w] : h[b * HID + (w - INSZ)];
        float half = 0.5f * ang;
        ch[w] = cosf(half);
        sh[w] = sinf(half);
    }

    // ---- initial product state: amp(idx) = (prod mags) * (-i)^popcount ----
    for (int t = 0; t < NSTATE / 256; ++t) {
        int idx = tid + 256 * t;
        float m = 1.0f;
#pragma unroll
        for (int w = 0; w < NQ; ++w) {
            int bit = (idx >> (NQ - 1 - w)) & 1;   // wire 0 = MSB
            m *= bit ? sh[w] : ch[w];
        }
        int k = __popc(idx) & 3;                   // (-i)^k
        float2 a;
        if      (k == 0) a = make_float2( m, 0.0f);
        else if (k == 1) a = make_float2(0.0f, -m);
        else if (k == 2) a = make_float2(-m, 0.0f);
        else             a = make_float2(0.0f,  m);
        st[idx] = a;
    }
    __syncthreads();

    // ---- layers: 14 Rot gates then 14 CNOTs ----
    for (int l = 0; l < NLAYERS; ++l) {
        for (int w = 0; w < NQ; ++w) {
            const float* g = wts + (l * NQ + w) * 3;
            float phi = g[0], th = g[1], om = g[2];
            float c = cosf(0.5f * th), s = sinf(0.5f * th);
            float sa, ca, sb, cb;
            sincosf(0.5f * (phi - om), &sa, &ca);  // ep  = (ca,  sa)
            sincosf(0.5f * (phi + om), &sb, &cb);  // em  = (cb, -sb)
            float2 U00 = make_float2( c * cb, -c * sb);   //  em*c
            float2 U01 = make_float2(-s * ca, -s * sa);   // -ep*s
            float2 U10 = make_float2( s * ca, -s * sa);   //  conj(ep)*s
            float2 U11 = make_float2( c * cb,  c * sb);   //  conj(em)*c

            const int p     = NQ - 1 - w;
            const int pmask = (1 << p) - 1;
            for (int t = 0; t < NSTATE / 512; ++t) {       // 8192 pairs / 256 thr
                int pr = tid + 256 * t;
                int i0 = ((pr & ~pmask) << 1) | (pr & pmask);
                int i1 = i0 | (1 << p);
                float2 s0 = st[i0], s1 = st[i1];
                st[i0] = cadd(cmul(U00, s0), cmul(U01, s1));
                st[i1] = cadd(cmul(U10, s0), cmul(U11, s1));
            }
            __syncthreads();
        }

        const int r = (l % (NQ - 1)) + 1;
        for (int w = 0; w < NQ; ++w) {
            const int pc = NQ - 1 - w;                 // control bit position
            const int pt = NQ - 1 - ((w + r) % NQ);    // target bit position
            const int lo = pc < pt ? pc : pt;
            const int hi = pc < pt ? pt : pc;
            for (int t = 0; t < (NSTATE / 4) / 256; ++t) {  // 4096 swaps / 256 thr
                int v = tid + 256 * t;                      // 12-bit enumerator
                v = ((v & ~((1 << lo) - 1)) << 1) | (v & ((1 << lo) - 1));
                v = ((v & ~((1 << hi) - 1)) << 1) | (v & ((1 << hi) - 1));
                int i0 = v | (1 << pc);                     // ctrl=1, tgt=0
                int i1 = i0 | (1 << pt);                    // ctrl=1, tgt=1
                float2 a0 = st[i0], a1 = st[i1];
                st[i0] = a1;
                st[i1] = a0;
            }
            __syncthreads();
        }
    }

    // ---- Z expectations: exp_q = sum_idx |amp|^2 * (1 - 2*bit_q) ----
    float acc[NQ];
#pragma unroll
    for (int q = 0; q < NQ; ++q) acc[q] = 0.0f;
    for (int t = 0; t < NSTATE / 256; ++t) {
        int idx = tid + 256 * t;
        float2 a = st[idx];
        float p = a.x * a.x + a.y * a.y;
#pragma unroll
        for (int q = 0; q < NQ; ++q)
            acc[q] += ((idx >> (NQ - 1 - q)) & 1) ? -p : p;
    }

    if (tid < 16) red[tid] = 0.0f;
    __syncthreads();
    const int lane = tid & 31;                       // wave32
#pragma unroll
    for (int q = 0; q < NQ; ++q) {
        float v = acc[q];
        for (int off = 16; off > 0; off >>= 1)
            v += __shfl_xor(v, off, 32);
        if (lane == 0) atomicAdd(&red[q], v);        // ds_add_f32
    }
    __syncthreads();
    if (tid < 16)
        Qws[b * 16 + tid] = (tid < NQ) ? red[tid] : 0.0f;  // zero-pad K to 16
}

// =====================================================================
// Kernel 2: out = tanh(q @ fc_w^T + fc_b) via V_WMMA_F32_16X16X4_F32.
// 16 row tiles of 16; K padded 14->16 as 4 chained k=4 WMMAs; N padded
// 8->16 with zero columns. One wave (32 threads) per tile.
// =====================================================================
__global__ __launch_bounds__(32) void fc_wmma_kernel(
    const float* __restrict__ Qws,    // [256, 16]
    const float* __restrict__ fc_w,   // [8, 14]
    const float* __restrict__ fc_b,   // [8]
    float* __restrict__ out)          // [256, 8]
{
    const int tile = blockIdx.x;          // 0..15
    const int lane = threadIdx.x;         // 0..31
    const int m    = lane & 15;
    const int koff = (lane >> 4) << 1;    // lanes 0-15 -> K{0,1}; 16-31 -> K{2,3}

#if __has_builtin(__builtin_amdgcn_wmma_f32_16x16x4_f32)
    v8f c = {};
#pragma unroll
    for (int s = 0; s < 4; ++s) {
        const int kb = 4 * s + koff;
        // A (16x4 f32): lane = M row, 2 VGPRs = 2 consecutive K values
        v2f a;
        a.x = Qws[(tile * 16 + m) * 16 + kb];
        a.y = Qws[(tile * 16 + m) * 16 + kb + 1];
        // B (4x16 f32): lane = N col; B[k][n] = fc_w[n][k], zero-padded
        v2f bv;
        bv.x = (m < HID && kb     < NQ) ? fc_w[m * NQ + kb]     : 0.0f;
        bv.y = (m < HID && kb + 1 < NQ) ? fc_w[m * NQ + kb + 1] : 0.0f;
        c = __builtin_amdgcn_wmma_f32_16x16x4_f32(
                /*neg_a=*/false, a, /*neg_b=*/false, bv,
                /*c_mod=*/(short)0, c, /*reuse_a=*/false, /*reuse_b=*/false);
    }
    // D layout: VGPR v -> M = v + 8*(lane>=16), N = lane & 15
    if (m < HID) {
        const int mbase = tile * 16 + ((lane >> 4) << 3);
        const float bias = fc_b[m];
#pragma unroll
        for (int v = 0; v < 8; ++v)
            out[(mbase + v) * HID + m] = tanhf(c[v] + bias);
    }
#else
    // scalar fallback (kept compile-safe; histogram will show wmma=0)
    const int gid = tile * 32 + lane;
    for (int o = gid; o < BATCH * HID; o += 16 * 32) {
        int row = o >> 3, n = o & 7;
        float acc = fc_b[n];
        for (int k = 0; k < NQ; ++k)
            acc += Qws[row * 16 + k] * fc_w[n * NQ + k];
        out[o] = tanhf(acc);
    }
#endif
}

// =====================================================================
extern "C" void kernel_launch(void* const* d_in, const int* in_sizes, int n_in,
                              void* d_out, int out_size, void* d_ws, size_t ws_size,
                              hipStream_t stream) {
    const float* x     = (const float*)d_in[0];   // [256,6]
    const float* h     = (const float*)d_in[1];   // [256,8]
    const float* wts   = (const float*)d_in[2];   // [2,14,3]
    const float* fc_w  = (const float*)d_in[3];   // [8,14]
    const float* fc_b  = (const float*)d_in[4];   // [8]
    float*       out   = (float*)d_out;           // [256,8]
    float*       Qws   = (float*)d_ws;            // [256,16] staged q (padded)

    const size_t lds_bytes = (size_t)NSTATE * sizeof(float2) + 16 * sizeof(float);

    qcircuit_kernel<<<BATCH, 256, lds_bytes, stream>>>(x, h, wts, Qws);
    fc_wmma_kernel<<<BATCH / 16, 32, 0, stream>>>(Qws, fc_w, fc_b, out);
}